// GCN_38302518345827
// MI455X (gfx1250) — compile-verified
//
#include <hip/hip_runtime.h>

// ---------------------------------------------------------------------------
// GraphSAGE (3 layers, mean aggr) + global mean pool + linear head for gfx1250.
// Dense GEMMs use V_WMMA_F32_16X16X4_F32 (fp32 matrix pipe, keeps reference
// precision). Sparse phases use hardware fp32 global atomics.
// ---------------------------------------------------------------------------

#define N_NODES 50000
#define N_EDGES 600000
#define HDIM    128
#define NGRAPH  64
#define NOUT    4

typedef __attribute__((ext_vector_type(2))) float v2f;
typedef __attribute__((ext_vector_type(8))) float v8f;

// ---------------------------------------------------------------- utilities

__global__ void gcn_zero_f32(float* __restrict__ p, int n) {
    int i = blockIdx.x * blockDim.x + threadIdx.x;
    if (i < n) p[i] = 0.0f;
}

// in-degree counts (shared by all 3 layers: dst ids never change)
__global__ void gcn_count_edges(const int* __restrict__ dst,
                                float* __restrict__ cnt, int e) {
    int i = blockIdx.x * blockDim.x + threadIdx.x;
    if (i < e) unsafeAtomicAdd(&cnt[dst[i]], 1.0f);
}

// agg[dst] += feat[src]; one thread per (edge, 4-float chunk): E*32 threads
__global__ void gcn_scatter_sum(const float* __restrict__ feat,
                                const int* __restrict__ src,
                                const int* __restrict__ dst,
                                float* __restrict__ agg, int e) {
    int idx  = blockIdx.x * blockDim.x + threadIdx.x;
    int edge = idx >> 5;
    if (edge >= e) return;
    int base = (idx & 31) * 4;
    const float4 v = *(const float4*)(feat + (size_t)src[edge] * HDIM + base);
    float* d = agg + (size_t)dst[edge] * HDIM + base;
    unsafeAtomicAdd(d + 0, v.x);
    unsafeAtomicAdd(d + 1, v.y);
    unsafeAtomicAdd(d + 2, v.z);
    unsafeAtomicAdd(d + 3, v.w);
}

// agg[n][h] /= max(count[n], 1)
__global__ void gcn_div_cnt(float* __restrict__ agg,
                            const float* __restrict__ cnt, int total) {
    int i = blockIdx.x * blockDim.x + threadIdx.x;
    if (i < total) agg[i] = agg[i] / fmaxf(cnt[i >> 7], 1.0f);
}

// ------------------------------------------------------------- WMMA GEMM
// out[16-row tile] = agg @ Wl^T + bl + xin @ Wr^T   (optionally ReLU)
// block = 256 threads = 8 waves; wave w owns 16-column tile w.
// A fragments staged via LDS (coalesced float4 fill, ds_load_b64 reads).
// Fragment layout (ISA 7.12.2, f32 16x16x4):
//   A/B: lane L -> row/col = L%16, VGPR v -> K = v + 2*(L/16)
//   C/D: lane L -> col = L%16, VGPR r -> row = r + 8*(L/16)
__global__ void gcn_sage_wmma(const float* __restrict__ agg,
                              const float* __restrict__ xin,
                              const float* __restrict__ Wl,
                              const float* __restrict__ bl,
                              const float* __restrict__ Wr,
                              float* __restrict__ out, int do_relu) {
    __shared__ float sA[16 * HDIM];   // agg tile
    __shared__ float sX[16 * HDIM];   // x   tile

    const int t     = threadIdx.x;
    const int lane  = t & 31;
    const int wave  = t >> 5;          // 0..7 -> column tile
    const int row0  = blockIdx.x * 16; // 3125 blocks * 16 = 50000 (exact)
    const int col0  = wave * 16;
    const int mrow  = lane & 15;
    const int khalf = (lane >> 4) * 2; // 0 or 2

    // cooperative coalesced staging: 2048 floats each = 512 float4, 256 thr
    {
        const float4* ga = (const float4*)(agg + (size_t)row0 * HDIM);
        const float4* gx = (const float4*)(xin + (size_t)row0 * HDIM);
        float4* la = (float4*)sA;
        float4* lx = (float4*)sX;
        la[t]       = ga[t];
        la[t + 256] = ga[t + 256];
        lx[t]       = gx[t];
        lx[t + 256] = gx[t + 256];
    }
    __syncthreads();

    const float* wlrow = Wl + (size_t)(col0 + mrow) * HDIM; // B col = mrow
    const float* wrrow = Wr + (size_t)(col0 + mrow) * HDIM;
    const float* arow  = sA + mrow * HDIM;
    const float* xrow  = sX + mrow * HDIM;

    v8f c = {};
    #pragma unroll
    for (int k = 0; k < HDIM; k += 4) {
        v2f a = *(const v2f*)(arow + k + khalf);   // ds_load_b64
        v2f b = *(const v2f*)(wlrow + k + khalf);  // global (L2-resident)
        c = __builtin_amdgcn_wmma_f32_16x16x4_f32(
                false, a, false, b, (short)0, c, false, false);
    }
    #pragma unroll
    for (int k = 0; k < HDIM; k += 4) {
        v2f a = *(const v2f*)(xrow + k + khalf);
        v2f b = *(const v2f*)(wrrow + k + khalf);
        c = __builtin_amdgcn_wmma_f32_16x16x4_f32(
                false, a, false, b, (short)0, c, false, false);
    }

    // write-back: fuse bias (+ReLU)
    const int col = lane & 15;
    const float bias = bl[col0 + col];
    #pragma unroll
    for (int r = 0; r < 8; ++r) {
        int m = r + 8 * (lane >> 4);
        float v = c[r] + bias;
        if (do_relu) v = fmaxf(v, 0.0f);
        out[(size_t)(row0 + m) * HDIM + col0 + col] = v;
    }
}

// ------------------------------------------------------------- pooling/head

__global__ void gcn_pool_sum(const float* __restrict__ h,
                             const int* __restrict__ batch,
                             float* __restrict__ gsum, int total) {
    int i = blockIdx.x * blockDim.x + threadIdx.x;
    if (i >= total) return;
    int n = i >> 7, hh = i & 127;
    unsafeAtomicAdd(&gsum[(size_t)batch[n] * HDIM + hh], h[i]);
}

__global__ void gcn_pool_cnt(const int* __restrict__ batch,
                             float* __restrict__ gcnt, int n) {
    int i = blockIdx.x * blockDim.x + threadIdx.x;
    if (i < n) unsafeAtomicAdd(&gcnt[batch[i]], 1.0f);
}

// out[g][o] = (1/max(cnt,1)) * <gsum[g], Wlin[o]> + blin[o]; 64*4 = 256 thr
__global__ void gcn_final_lin(const float* __restrict__ gsum,
                              const float* __restrict__ gcnt,
                              const float* __restrict__ Wlin,
                              const float* __restrict__ blin,
                              float* __restrict__ out) {
    int t = threadIdx.x;           // 0..255
    int g = t >> 2, o = t & 3;
    float inv = 1.0f / fmaxf(gcnt[g], 1.0f);
    const float* gs = gsum + (size_t)g * HDIM;
    const float* w  = Wlin + (size_t)o * HDIM;
    float acc = 0.0f;
    #pragma unroll 8
    for (int h = 0; h < HDIM; ++h) acc += gs[h] * w[h];
    out[g * NOUT + o] = acc * inv + blin[o];
}

// ------------------------------------------------------------------ driver

extern "C" void kernel_launch(void* const* d_in, const int* in_sizes, int n_in,
                              void* d_out, int out_size, void* d_ws, size_t ws_size,
                              hipStream_t stream) {
    const float* x    = (const float*)d_in[0];
    const int*   ei   = (const int*)d_in[1];
    const int*   src  = ei;
    const int*   dst  = ei + N_EDGES;
    const int*   batch = (const int*)d_in[2];
    const float* Wl1 = (const float*)d_in[3];
    const float* bl1 = (const float*)d_in[4];
    const float* Wr1 = (const float*)d_in[5];
    const float* Wl2 = (const float*)d_in[6];
    const float* bl2 = (const float*)d_in[7];
    const float* Wr2 = (const float*)d_in[8];
    const float* Wl3 = (const float*)d_in[9];
    const float* bl3 = (const float*)d_in[10];
    const float* Wr3 = (const float*)d_in[11];
    const float* Wlin = (const float*)d_in[12];
    const float* blin = (const float*)d_in[13];
    float* out = (float*)d_out;

    float* ws   = (float*)d_ws;
    float* agg  = ws;                                  // N*H
    float* hA   = agg + (size_t)N_NODES * HDIM;        // N*H
    float* hB   = hA  + (size_t)N_NODES * HDIM;        // N*H
    float* cnt  = hB  + (size_t)N_NODES * HDIM;        // N
    float* gsum = cnt + N_NODES;                       // G*H
    float* gcnt = gsum + NGRAPH * HDIM;                // G  (contiguous w/ gsum)

    const int TPB = 256;
    const int NH       = N_NODES * HDIM;               // 6,400,000
    const int blkN     = (N_NODES + TPB - 1) / TPB;    // 196
    const int blkE     = (N_EDGES + TPB - 1) / TPB;    // 2344
    const int blkNH    = NH / TPB;                     // 25000
    const int blkScat  = (N_EDGES * 32) / TPB;         // 75000
    const int blkGemm  = N_NODES / 16;                 // 3125
    const int blkPoolZ = (NGRAPH * HDIM + NGRAPH + TPB - 1) / TPB;

    // in-degree counts (identical for all layers)
    gcn_zero_f32<<<blkN, TPB, 0, stream>>>(cnt, N_NODES);
    gcn_count_edges<<<blkE, TPB, 0, stream>>>(dst, cnt, N_EDGES);

    // ---- layer 1: x -> hA (ReLU)
    gcn_zero_f32<<<blkNH, TPB, 0, stream>>>(agg, NH);
    gcn_scatter_sum<<<blkScat, TPB, 0, stream>>>(x, src, dst, agg, N_EDGES);
    gcn_div_cnt<<<blkNH, TPB, 0, stream>>>(agg, cnt, NH);
    gcn_sage_wmma<<<blkGemm, TPB, 0, stream>>>(agg, x, Wl1, bl1, Wr1, hA, 1);

    // ---- layer 2: hA -> hB (ReLU)
    gcn_zero_f32<<<blkNH, TPB, 0, stream>>>(agg, NH);
    gcn_scatter_sum<<<blkScat, TPB, 0, stream>>>(hA, src, dst, agg, N_EDGES);
    gcn_div_cnt<<<blkNH, TPB, 0, stream>>>(agg, cnt, NH);
    gcn_sage_wmma<<<blkGemm, TPB, 0, stream>>>(agg, hA, Wl2, bl2, Wr2, hB, 1);

    // ---- layer 3: hB -> hA (no ReLU)
    gcn_zero_f32<<<blkNH, TPB, 0, stream>>>(agg, NH);
    gcn_scatter_sum<<<blkScat, TPB, 0, stream>>>(hB, src, dst, agg, N_EDGES);
    gcn_div_cnt<<<blkNH, TPB, 0, stream>>>(agg, cnt, NH);
    gcn_sage_wmma<<<blkGemm, TPB, 0, stream>>>(agg, hB, Wl3, bl3, Wr3, hA, 0);

    // ---- global mean pool + linear head
    gcn_zero_f32<<<blkPoolZ, TPB, 0, stream>>>(gsum, NGRAPH * HDIM + NGRAPH);
    gcn_pool_sum<<<blkNH, TPB, 0, stream>>>(hA, batch, gsum, NH);
    gcn_pool_cnt<<<blkN, TPB, 0, stream>>>(batch, gcnt, N_NODES);
    gcn_final_lin<<<1, TPB, 0, stream>>>(gsum, gcnt, Wlin, blin, out);
}